// GAT_25958782337775
// MI455X (gfx1250) — compile-verified
//
#include <hip/hip_runtime.h>
#include <math.h>

typedef __attribute__((ext_vector_type(2))) float v2f;
typedef __attribute__((ext_vector_type(8))) float v8f;

#define N_NODES   50000
#define E_RAW     800000
#define E_TOT     (E_RAW + N_NODES)   // self loops appended
#define DIN       512
#define H1        8
#define D1        16
#define HD1       128
#define D2        64
#define NEG_SLOPE 0.2f
#define GAT_EPS   1e-16f

// ---------- float <-> order-preserving uint (for atomic float max) ----------
__device__ __forceinline__ unsigned f2ord(float f) {
  unsigned u = __float_as_uint(f);
  return (u & 0x80000000u) ? ~u : (u | 0x80000000u);
}
__device__ __forceinline__ float ord2f(unsigned u) {
  unsigned v = (u & 0x80000000u) ? (u ^ 0x80000000u) : ~u;
  return __uint_as_float(v);
}

// ---------------------------------------------------------------------------
// Fused dual GEMM: C0 = A @ B0, C1 = A @ B1  (A read ONCE — halves the
// dominant HBM stream, x is 102.4 MB). fp32, one wave per 16x16 tile pair,
// V_WMMA_F32_16X16X4_F32. Requires M%16==0, N%16==0, K%4==0.
// A-layout: lanes 0-15 -> K=k,k+1 ; lanes 16-31 -> K=k+2,k+3 (ISA 16x4 f32)
// B-layout: lane halves split K contiguously (mirror of A)
// C-layout: VGPR j -> row j + (lane>=16)*8, col lane&15
// ---------------------------------------------------------------------------
__global__ void __launch_bounds__(32)
gemm2_f32_wmma(const float* __restrict__ A,
               const float* __restrict__ B0, const float* __restrict__ B1,
               float* __restrict__ C0, float* __restrict__ C1,
               int M, int N, int K) {
  const int lane = threadIdx.x;
  const int half = lane >> 4;     // 0 or 1
  const int l    = lane & 15;
  const int row0 = blockIdx.x * 16;
  const int col0 = blockIdx.y * 16;

  const float* __restrict__ arow  = A  + (size_t)(row0 + l) * K;
  const float* __restrict__ b0col = B0 + col0 + l;
  const float* __restrict__ b1col = B1 + col0 + l;

  v8f acc0 = {};
  v8f acc1 = {};
  for (int k = 0; k < K; k += 4) {
    const int ka = k + half * 2;            // even -> 8B-aligned float2 load
    const float2 av = *reinterpret_cast<const float2*>(arow + ka);
    v2f a;  a.x = av.x;           a.y = av.y;
    v2f b0; b0.x = b0col[(size_t)ka * N]; b0.y = b0col[(size_t)(ka + 1) * N];
    v2f b1; b1.x = b1col[(size_t)ka * N]; b1.y = b1col[(size_t)(ka + 1) * N];
    acc0 = __builtin_amdgcn_wmma_f32_16x16x4_f32(false, a, false, b0,
                                                 (short)0, acc0, false, false);
    acc1 = __builtin_amdgcn_wmma_f32_16x16x4_f32(false, a, false, b1,
                                                 (short)0, acc1, false, false);
  }

  float* __restrict__ c0row = C0 + (size_t)(row0 + half * 8) * N + col0 + l;
  float* __restrict__ c1row = C1 + (size_t)(row0 + half * 8) * N + col0 + l;
#pragma unroll
  for (int j = 0; j < 8; ++j) {
    c0row[(size_t)j * N] = acc0[j];
    c1row[(size_t)j * N] = acc1[j];
  }
}

// ---------------------------------------------------------------------------
// Pass A: per (edge, head) GATv2 logit + segment max (ordered-uint atomicMax)
// ---------------------------------------------------------------------------
__global__ void edge_logits_max(const int* __restrict__ esrc,
                                const int* __restrict__ edst,
                                const float* __restrict__ XL,
                                const float* __restrict__ XR,
                                const float* __restrict__ att,
                                float* __restrict__ logits,
                                unsigned* __restrict__ menc,
                                int H, int D) {
  const int t = blockIdx.x * blockDim.x + threadIdx.x;
  if (t >= E_TOT * H) return;
  const int e = t / H;
  const int h = t - e * H;
  const int s = (e < E_RAW) ? esrc[e] : (e - E_RAW);
  const int d = (e < E_RAW) ? edst[e] : (e - E_RAW);

  const float* __restrict__ xl = XL + ((size_t)s * H + h) * D;
  const float* __restrict__ xr = XR + ((size_t)d * H + h) * D;
  const float* __restrict__ a  = att + (size_t)h * D;

  float acc = 0.f;
  for (int i = 0; i < D; ++i) {
    float v = xl[i] + xr[i];
    v = (v > 0.f) ? v : NEG_SLOPE * v;   // leaky_relu
    acc = fmaf(a[i], v, acc);
  }
  logits[t] = acc;
  atomicMax(&menc[(size_t)d * H + h], f2ord(acc));
}

// ---------------------------------------------------------------------------
// Pass B: ex = exp(logit - max); segment sum of ex (global_atomic_add_f32)
// ---------------------------------------------------------------------------
__global__ void edge_exp_sum(const int* __restrict__ edst,
                             float* __restrict__ logits,
                             const unsigned* __restrict__ menc,
                             float* __restrict__ denom, int H) {
  const int t = blockIdx.x * blockDim.x + threadIdx.x;
  if (t >= E_TOT * H) return;
  const int e = t / H;
  const int h = t - e * H;
  const int d = (e < E_RAW) ? edst[e] : (e - E_RAW);

  const float m  = ord2f(menc[(size_t)d * H + h]);  // finite: self loops exist
  const float ex = expf(logits[t] - m);
  logits[t] = ex;                                    // store ex in place
  atomicAdd(&denom[(size_t)d * H + h], ex);
}

// ---------------------------------------------------------------------------
// Pass C: alpha = ex / (denom + eps); out[dst] += alpha * x_l[src]
// ---------------------------------------------------------------------------
__global__ void edge_aggregate(const int* __restrict__ esrc,
                               const int* __restrict__ edst,
                               const float* __restrict__ logits,  // holds ex
                               const float* __restrict__ denom,
                               const float* __restrict__ XL,
                               float* __restrict__ OUT, int H, int D) {
  const int t = blockIdx.x * blockDim.x + threadIdx.x;
  if (t >= E_TOT * H) return;
  const int e = t / H;
  const int h = t - e * H;
  const int s = (e < E_RAW) ? esrc[e] : (e - E_RAW);
  const int d = (e < E_RAW) ? edst[e] : (e - E_RAW);

  const float alpha = logits[t] / (denom[(size_t)d * H + h] + GAT_EPS);
  const float* __restrict__ xl = XL + ((size_t)s * H + h) * D;
  float* __restrict__ o        = OUT + ((size_t)d * H + h) * D;
  for (int i = 0; i < D; ++i) atomicAdd(&o[i], alpha * xl[i]);
}

// ---------------------------------------------------------------------------
// h = elu(agg + b1), in place (layer-1 activation)
// ---------------------------------------------------------------------------
__global__ void bias_elu(float* __restrict__ buf, const float* __restrict__ b,
                         int total, int C) {
  const int t = blockIdx.x * blockDim.x + threadIdx.x;
  if (t >= total) return;
  const int c = t % C;
  const float v = buf[t] + b[c];
  buf[t] = (v > 0.f) ? v : (expf(v) - 1.f);
}

// ---------------------------------------------------------------------------
// out = log_softmax(row + b2) per node, in place on d_out (64 cols)
// ---------------------------------------------------------------------------
__global__ void bias_logsoftmax(float* __restrict__ out,
                                const float* __restrict__ b2) {
  const int n = blockIdx.x * blockDim.x + threadIdx.x;
  if (n >= N_NODES) return;
  float* __restrict__ row = out + (size_t)n * D2;
  float mx = -INFINITY;
  for (int i = 0; i < D2; ++i) mx = fmaxf(mx, row[i] + b2[i]);
  float se = 0.f;
  for (int i = 0; i < D2; ++i) se += expf(row[i] + b2[i] - mx);
  const float lse = mx + logf(se);
  for (int i = 0; i < D2; ++i) row[i] = row[i] + b2[i] - lse;
}

// ---------------------------------------------------------------------------
extern "C" void kernel_launch(void* const* d_in, const int* in_sizes, int n_in,
                              void* d_out, int out_size, void* d_ws, size_t ws_size,
                              hipStream_t stream) {
  const float* x    = (const float*)d_in[0];
  const int*   ei   = (const int*)d_in[1];      // [2, E_RAW]
  const float* Wl1  = (const float*)d_in[2];
  const float* Wr1  = (const float*)d_in[3];
  const float* att1 = (const float*)d_in[4];
  const float* b1   = (const float*)d_in[5];
  const float* Wl2  = (const float*)d_in[6];
  const float* Wr2  = (const float*)d_in[7];
  const float* att2 = (const float*)d_in[8];
  const float* b2   = (const float*)d_in[9];

  const int* esrc = ei;
  const int* edst = ei + E_RAW;

  // ---- workspace layout (floats) ----
  float* ws = (float*)d_ws;
  float*    xl1    = ws;                                // N*128
  float*    xr1    = xl1 + (size_t)N_NODES * HD1;       // N*128
  float*    agg1   = xr1 + (size_t)N_NODES * HD1;       // N*128 (becomes h)
  float*    logits = agg1 + (size_t)N_NODES * HD1;      // E_TOT*8 (reused L2)
  unsigned* menc1  = (unsigned*)(logits + (size_t)E_TOT * H1);  // N*8
  float*    denom1 = (float*)(menc1 + (size_t)N_NODES * H1);    // N*8
  unsigned* menc2  = (unsigned*)(denom1 + (size_t)N_NODES * H1);// N
  float*    denom2 = (float*)(menc2 + N_NODES);                  // N
  // layer-2 transformed features alias layer-1 buffers (free after edge phase)
  float*    xl2    = xl1;                                // N*64 fits in N*128
  float*    xr2    = xr1;
  float*    outf   = (float*)d_out;                      // agg2 accumulator

  // ---- zero accumulators (graph-capture-safe memset nodes) ----
  hipMemsetAsync(menc1,  0, (size_t)N_NODES * H1 * 4, stream);
  hipMemsetAsync(denom1, 0, (size_t)N_NODES * H1 * 4, stream);
  hipMemsetAsync(agg1,   0, (size_t)N_NODES * HD1 * 4, stream);
  hipMemsetAsync(menc2,  0, (size_t)N_NODES * 4, stream);
  hipMemsetAsync(denom2, 0, (size_t)N_NODES * 4, stream);
  hipMemsetAsync(outf,   0, (size_t)N_NODES * D2 * 4, stream);

  const int TPB = 256;
  const int eh1 = E_TOT * H1;                 // 6.8M
  const int eh2 = E_TOT;                      // 850K

  // ---- layer 1: fused dual WMMA GEMM (x read once) ----
  {
    dim3 grid(N_NODES / 16, HD1 / 16);        // 3125 x 8 waves
    gemm2_f32_wmma<<<grid, 32, 0, stream>>>(x, Wl1, Wr1, xl1, xr1,
                                            N_NODES, HD1, DIN);
  }

  // ---- layer 1: edge phase (H=8, D=16) ----
  edge_logits_max<<<(eh1 + TPB - 1) / TPB, TPB, 0, stream>>>(
      esrc, edst, xl1, xr1, att1, logits, menc1, H1, D1);
  edge_exp_sum<<<(eh1 + TPB - 1) / TPB, TPB, 0, stream>>>(
      edst, logits, menc1, denom1, H1);
  edge_aggregate<<<(eh1 + TPB - 1) / TPB, TPB, 0, stream>>>(
      esrc, edst, logits, denom1, xl1, agg1, H1, D1);

  // ---- h = elu(agg1 + b1) in place ----
  {
    const int tot = N_NODES * HD1;
    bias_elu<<<(tot + TPB - 1) / TPB, TPB, 0, stream>>>(agg1, b1, tot, HD1);
  }

  // ---- layer 2: fused dual WMMA GEMM (h read once) ----
  {
    dim3 grid(N_NODES / 16, D2 / 16);         // 3125 x 4 waves
    gemm2_f32_wmma<<<grid, 32, 0, stream>>>(agg1, Wl2, Wr2, xl2, xr2,
                                            N_NODES, D2, HD1);
  }

  // ---- layer 2: edge phase (H=1, D=64), accumulate into d_out ----
  edge_logits_max<<<(eh2 + TPB - 1) / TPB, TPB, 0, stream>>>(
      esrc, edst, xl2, xr2, att2, logits, menc2, 1, D2);
  edge_exp_sum<<<(eh2 + TPB - 1) / TPB, TPB, 0, stream>>>(
      edst, logits, menc2, denom2, 1);
  edge_aggregate<<<(eh2 + TPB - 1) / TPB, TPB, 0, stream>>>(
      esrc, edst, logits, denom2, xl2, outf, 1, D2);

  // ---- final: log_softmax(out + b2) ----
  bias_logsoftmax<<<(N_NODES + TPB - 1) / TPB, TPB, 0, stream>>>(outf, b2);
}